// PolyLinear_40218073760341
// MI455X (gfx1250) — compile-verified
//
#include <hip/hip_runtime.h>
#include <hip/hip_bf16.h>

// ---------------------------------------------------------------------------
// PolyLinear: out[b] = bias + sum_t W_t * xa_i xa_j xa_k   (padded triples)
// Reformulated as U = Z2 @ Wmat (f32 WMMA 16x16x4), out[b] = bias + <x_b, U_b>
//   Z2[b, p(j,k)] = xa_j*xa_k over 561 canonical pairs, Wmat is 561x32 scatter.
// Each wave computes an M=32 strip (two 16-row tiles) so B fragments and the
// pair table are amortized over 4 WMMAs per K-step.
// ---------------------------------------------------------------------------

#define NF        32          // features
#define NA        33          // augmented (x, 1); also xs row stride (odd -> 16 banks)
#define NPAIR     561         // 33*34/2 canonical (j<=k) pairs
#define KP        564         // NPAIR padded to multiple of 4 (WMMA K-step)
#define BSTRIDE   40          // bs row stride: +2 rows = +80 dw = +16 banks (conflict-free)
#define ROWS_WG   128         // 4 waves x 32 rows
#define THREADS   128

typedef __attribute__((ext_vector_type(2))) float v2f;
typedef __attribute__((ext_vector_type(8))) float v8f;

// --- kernel 1: zero padded Wmat (KP x 32) ----------------------------------
__global__ void poly_zero_wmat(float* __restrict__ wm, int n) {
    int i = blockIdx.x * blockDim.x + threadIdx.x;
    if (i < n) wm[i] = 0.0f;
}

// --- kernel 2: scatter W into Wmat[p(j,k)*32 + i] --------------------------
__global__ void poly_scatter_wmat(const int* __restrict__ idx,
                                  const float* __restrict__ W,
                                  float* __restrict__ wm, int nTerms) {
    int t = blockIdx.x * blockDim.x + threadIdx.x;
    if (t >= nTerms) return;
    int i = idx[3 * t + 0];          // 0..31  (sorted ascending, degree>=1)
    int j = idx[3 * t + 1];          // 0..32
    int k = idx[3 * t + 2];          // j..32
    int p = (k * (k + 1)) / 2 + j;   // canonical pair index, 0..560
    wm[p * NF + i] = W[t];           // injective mapping -> plain store
}

// --- kernel 3: fused Z2-build + WMMA GEMM + row-dot epilogue ---------------
__global__ void __launch_bounds__(THREADS)
poly_main(const float* __restrict__ x,
          const float* __restrict__ bias_p,
          const float* __restrict__ wm,
          float* __restrict__ out, int batch) {
    // dynamic LDS layout:
    //   xs   : ROWS_WG x NA floats (stride 33), augmented rows
    //   bs   : KP x BSTRIDE floats (cols 0..31 used), staged Wmat
    //   jk16 : KP u16, packed (k<<8)|j per pair (loaded pairwise as u32)
    extern __shared__ float smem[];
    float*          xs   = smem;                           // ROWS_WG*NA
    float*          bs   = smem + ROWS_WG * NA;            // KP*BSTRIDE
    unsigned short* jk16 = (unsigned short*)(bs + KP * BSTRIDE);

    const int tid     = threadIdx.x;
    const int lane    = tid & 31;
    const int wave    = tid >> 5;
    const int half    = (lane >> 4);     // 0: K rows +0/+1, 1: K rows +2/+3
    const int nIdx    = lane & 15;       // N column within 16-wide tile
    const int rowBase = blockIdx.x * ROWS_WG;

    // stage Wmat (global stride 32 -> LDS stride 40)
    for (int i = tid; i < KP * NF; i += THREADS) {
        int p = i >> 5, c = i & 31;
        bs[p * BSTRIDE + c] = wm[i];
    }
    // stage 128 augmented rows
    for (int i = tid; i < ROWS_WG * NF; i += THREADS) {
        int r = i >> 5, c = i & 31;
        int g = rowBase + r;
        xs[r * NA + c] = (g < batch) ? x[g * NF + c] : 0.0f;
    }
    if (tid < ROWS_WG) xs[tid * NA + NF] = 1.0f;   // ones column
    // pair decode table: p -> (j,k), j<=k, clamped for padded tail
    for (int p = tid; p < KP; p += THREADS) {
        int k = (int)((__builtin_sqrtf(8.0f * (float)p + 1.0f) - 1.0f) * 0.5f);
        while ((k + 1) * (k + 2) / 2 <= p) ++k;
        while (k * (k + 1) / 2 > p) --k;
        int j = p - k * (k + 1) / 2;
        if (k > 32) k = 32;
        if (j > 32) j = 32;
        jk16[p] = (unsigned short)((k << 8) | j);
    }
    __syncthreads();

    // A-matrix rows this lane feeds (16x4 f32 A layout: lanes L and L+16 share M=L&15)
    const int xb0 = (wave * 32 + nIdx) * NA;       // tile 0 (rows 0..15 of strip)
    const int xb1 = xb0 + 16 * NA;                 // tile 1 (rows 16..31 of strip)

    v8f acc00 = {0.f,0.f,0.f,0.f,0.f,0.f,0.f,0.f}; // tile0 x cols 0..15
    v8f acc01 = {0.f,0.f,0.f,0.f,0.f,0.f,0.f,0.f}; // tile0 x cols 16..31
    v8f acc10 = {0.f,0.f,0.f,0.f,0.f,0.f,0.f,0.f}; // tile1 x cols 0..15
    v8f acc11 = {0.f,0.f,0.f,0.f,0.f,0.f,0.f,0.f}; // tile1 x cols 16..31

    // uniform trip count, no divergence: EXEC all-ones around WMMA
    for (int kk = 0; kk < KP; kk += 4) {
        const int p0 = kk + half * 2;              // even; this lane's VGPR0 K-row
        // one aligned 32-bit DS load fetches jk16[p0] (lo) and jk16[p0+1] (hi)
        const unsigned int e = *(const unsigned int*)&jk16[p0];
        const int j0 =  e        & 0xffu;
        const int k0 = (e >> 8)  & 0xffu;
        const int j1 = (e >> 16) & 0xffu;
        const int k1 =  e >> 24;
        // A fragments: Z2[row, p] = xa_j * xa_k  (B rows >= NPAIR are zero,
        // so the padded tail contributes nothing regardless of A)
        v2f A0 = { xs[xb0 + j0] * xs[xb0 + k0], xs[xb0 + j1] * xs[xb0 + k1] };
        v2f A1 = { xs[xb1 + j0] * xs[xb1 + k0], xs[xb1 + j1] * xs[xb1 + k1] };
        // B fragments via one base pointer -> ds_load_2addr pairs
        const float* bp = &bs[p0 * BSTRIDE + nIdx];
        v2f B0 = { bp[0],  bp[BSTRIDE]      };     // cols 0..15, K rows p0/p0+1
        v2f B1 = { bp[16], bp[BSTRIDE + 16] };     // cols 16..31
        acc00 = __builtin_amdgcn_wmma_f32_16x16x4_f32(false, A0, false, B0,
                                                      (short)0, acc00, false, false);
        acc01 = __builtin_amdgcn_wmma_f32_16x16x4_f32(false, A0, false, B1,
                                                      (short)0, acc01, false, false);
        acc10 = __builtin_amdgcn_wmma_f32_16x16x4_f32(false, A1, false, B0,
                                                      (short)0, acc10, false, false);
        acc11 = __builtin_amdgcn_wmma_f32_16x16x4_f32(false, A1, false, B1,
                                                      (short)0, acc11, false, false);
    }

    // Epilogue: out[row] = bias + sum_i x[row,i] * U[row,i]
    // C/D layout: VGPR v holds M = v + half*8, N = nIdx (acc*1 -> N+16)
    const float bias = bias_p[0];
#pragma unroll
    for (int t = 0; t < 2; ++t) {
        const v8f aN0 = t ? acc10 : acc00;
        const v8f aN1 = t ? acc11 : acc01;
        float pr[8];
#pragma unroll
        for (int v = 0; v < 8; ++v) {
            int row = wave * 32 + t * 16 + half * 8 + v;
            pr[v] = aN0[v] * xs[row * NA + nIdx] +
                    aN1[v] * xs[row * NA + 16 + nIdx];
        }
        // reduce across the 16 lanes of each half (xor<16 stays inside the half)
#pragma unroll
        for (int off = 1; off < 16; off <<= 1) {
#pragma unroll
            for (int v = 0; v < 8; ++v) pr[v] += __shfl_xor(pr[v], off, 32);
        }
        if (nIdx == 0) {
#pragma unroll
            for (int v = 0; v < 8; ++v) {
                int row = rowBase + wave * 32 + t * 16 + half * 8 + v;
                if (row < batch) out[row] = pr[v] + bias;
            }
        }
    }
}

// ---------------------------------------------------------------------------
extern "C" void kernel_launch(void* const* d_in, const int* in_sizes, int n_in,
                              void* d_out, int out_size, void* d_ws, size_t ws_size,
                              hipStream_t stream) {
    const float* x   = (const float*)d_in[0];
    const float* W   = (const float*)d_in[1];
    const float* b   = (const float*)d_in[2];
    const int*   idx = (const int*)d_in[3];
    float*       out = (float*)d_out;

    const int nTerms = in_sizes[1];          // 6544
    const int batch  = in_sizes[0] / NF;     // 32768

    float* wm = (float*)d_ws;                // KP*32 floats = 72192 B scratch

    const int wmElems = KP * NF;
    poly_zero_wmat<<<(wmElems + 255) / 256, 256, 0, stream>>>(wm, wmElems);
    poly_scatter_wmat<<<(nTerms + 255) / 256, 256, 0, stream>>>(idx, W, wm, nTerms);

    const size_t shBytes = (size_t)(ROWS_WG * NA + KP * BSTRIDE) * sizeof(float)
                         + (size_t)KP * sizeof(unsigned short);
    const int nBlocks = (batch + ROWS_WG - 1) / ROWS_WG;
    poly_main<<<nBlocks, THREADS, shBytes, stream>>>(x, b, wm, out, batch);
}